// BahdanauAttention_12695923327239
// MI455X (gfx1250) — compile-verified
//
#include <hip/hip_runtime.h>
#include <hip/hip_bf16.h>

typedef __attribute__((ext_vector_type(2))) float v2f;
typedef __attribute__((ext_vector_type(8))) float v8f;
typedef __attribute__((ext_vector_type(4))) int   v4i;

// ---------- CDNA5 async global->LDS copy (ASYNCcnt path), with sync fallback ----------
#if defined(__has_builtin)
#if __has_builtin(__builtin_amdgcn_global_load_async_to_lds_b128)
#define HAVE_ASYNC_LDS 1
#endif
#endif
#ifndef HAVE_ASYNC_LDS
#define HAVE_ASYNC_LDS 0
#endif

__device__ __forceinline__ void copy16_g2l(float* ldst, const float* gsrc) {
#if HAVE_ASYNC_LDS
    // emits global_load_async_to_lds_b128 (tracked by ASYNCcnt)
    __builtin_amdgcn_global_load_async_to_lds_b128(
        (__attribute__((address_space(1))) v4i*)gsrc,
        (__attribute__((address_space(3))) v4i*)ldst,
        0, 0);
#else
    *(float4*)ldst = *(const float4*)gsrc;
#endif
}

__device__ __forceinline__ void async_join() {
#if HAVE_ASYNC_LDS
    asm volatile("s_wait_asynccnt 0" ::: "memory");
#endif
}

// ---------- fast transcendentals (gfx1250 has v_tanh_f32) ----------
__device__ __forceinline__ float fast_exp(float x) {
#if __has_builtin(__builtin_amdgcn_exp2f)
    return __builtin_amdgcn_exp2f(x * 1.44269504088896340736f); // v_exp_f32
#else
    return __expf(x);
#endif
}

__device__ __forceinline__ float fast_tanh(float x) {
#if __has_builtin(__builtin_amdgcn_tanhf)
    return __builtin_amdgcn_tanhf(x);                 // v_tanh_f32
#elif __has_builtin(__builtin_amdgcn_tanh_f32)
    return __builtin_amdgcn_tanh_f32(x);
#else
    float e2x = fast_exp(2.0f * x);                   // tanh = 1 - 2/(e^2x+1)
    return 1.0f - 2.0f / (e2x + 1.0f);
#endif
}

// ---------- kernel 0: v = linear_att / ||linear_att|| * scalar ----------
__global__ void compute_v_kernel(const float* __restrict__ la,
                                 const float* __restrict__ scalar,
                                 float* __restrict__ v) {
    __shared__ float red[16];
    const int tid = threadIdx.x;          // 512 threads = 16 waves
    float x  = la[tid];
    float sq = x * x;
    #pragma unroll
    for (int off = 16; off; off >>= 1) sq += __shfl_xor(sq, off, 32);
    if ((tid & 31) == 0) red[tid >> 5] = sq;
    __syncthreads();
    if (tid < 16) {
        float s = red[tid];
        #pragma unroll
        for (int off = 8; off; off >>= 1) s += __shfl_xor(s, off, 32);
        if (tid == 0) red[0] = s;
    }
    __syncthreads();
    v[tid] = x * rsqrtf(red[0]) * scalar[0];
}

// ---------- kernel 1: Out[R][512] = X[R][512] * W^T   (W row-major [u][d]) ----------
// Block = 8 waves; block covers one 16-row strip x 128 u-cols (8 tiles of 16x16).
// Both A (16x128) and B (128x128) K-chunks staged in LDS with coalesced/async loads.
// Padded stride 132 (mod 64 = 4) -> conflict-free WMMA fragment reads.
// f32 WMMA 16x16x4 A/B frag: lane L -> row m=L%16, kbase=(L/16)*2, elems K=kbase,kbase+1
#define GEMM_KC 128
__global__ void gemm_xWt_wmma(const float* __restrict__ X,
                              const float* __restrict__ W,
                              float* __restrict__ Out) {
    __shared__ __align__(16) float As[16  * 132];
    __shared__ __align__(16) float Bs[128 * 132];
    const int tid  = threadIdx.x;
    const int wave = tid >> 5;
    const int lane = tid & 31;
    const int rowTile = blockIdx.x >> 2;          // 4 blocks per 16-row strip
    const int colGrp  = blockIdx.x & 3;
    const int rowBase = rowTile * 16;
    const int uBlock  = colGrp * 128;             // block's 128 u-rows of W
    const int uBase   = uBlock + wave * 16;       // wave's 16 u-rows
    const int m     = lane & 15;
    const int kbase = (lane >> 4) << 1;
    v8f acc = {};
    for (int k0 = 0; k0 < 512; k0 += GEMM_KC) {
        __syncthreads();   // all waves done reading LDS from previous chunk
        // A strip: 16 rows x 128 floats = 512 float4 slots
        for (int s = tid; s < 512; s += 256) {
            const int r = s >> 5, c4 = s & 31;    // 32 float4 per row
            copy16_g2l(&As[r * 132 + c4 * 4],
                       X + (size_t)(rowBase + r) * 512 + k0 + c4 * 4);
        }
        // B strip: 128 u-rows x 128 floats = 4096 float4 slots (coalesced along d)
        for (int s = tid; s < 4096; s += 256) {
            const int r = s >> 5, c4 = s & 31;
            copy16_g2l(&Bs[r * 132 + c4 * 4],
                       W + (size_t)(uBlock + r) * 512 + k0 + c4 * 4);
        }
        async_join();
        __syncthreads();
        const float* arow = &As[m * 132];
        const float* brow = &Bs[(wave * 16 + m) * 132];
        #pragma unroll
        for (int kk = 0; kk < GEMM_KC; kk += 4) {
            v2f a     = *(const v2f*)(arow + kk + kbase);
            v2f bfrag = *(const v2f*)(brow + kk + kbase);
            acc = __builtin_amdgcn_wmma_f32_16x16x4_f32(
                false, a, false, bfrag, (short)0, acc, false, false);
        }
    }
    // C/D layout: VGPR j -> M=j (lanes 0-15) / M=j+8 (lanes 16-31), N=lane%16
    const int col  = lane & 15;
    const int rsel = (lane >> 4) * 8;
    #pragma unroll
    for (int j = 0; j < 8; ++j)
        Out[(size_t)(rowBase + rsel + j) * 512 + uBase + col] = acc[j];
}

// ---------- kernel 2: scores + softmax (TRANS-bound hot loop) ----------
// grid = B*Q blocks; block = 256 threads = 8 waves; wave w owns k in [w*64, w*64+64)
__global__ void scores_softmax_kernel(const float* __restrict__ pq,
                                      const float* __restrict__ pk,
                                      const float* __restrict__ bias,
                                      const float* __restrict__ v,
                                      float* __restrict__ rawS,
                                      float* __restrict__ outNorm) {
    __shared__ float pqb[512];
    __shared__ float vsh[512];
    __shared__ float ssh[512];
    __shared__ float red[8];
    const int tid  = threadIdx.x;
    const int wave = tid >> 5;
    const int lane = tid & 31;
    const int bq   = blockIdx.x;       // 0..511
    const int b    = bq >> 6;
    const float* pqr = pq + (size_t)bq * 512;
    for (int i = tid; i < 512; i += 256) {
        pqb[i] = pqr[i] + bias[i];
        vsh[i] = v[i];
    }
    __syncthreads();
    const float* pkb = pk + (size_t)b * 512 * 512;
    for (int kk = 0; kk < 64; ++kk) {
        const int k = wave * 64 + kk;
        const float* pkr = pkb + (size_t)k * 512;
        float acc = 0.0f;
        #pragma unroll
        for (int ui = 0; ui < 16; ++ui) {
            const int u = lane + ui * 32;              // coalesced over lanes
            acc += fast_tanh(pqb[u] + pkr[u]) * vsh[u];
        }
        #pragma unroll
        for (int off = 16; off; off >>= 1) acc += __shfl_xor(acc, off, 32);
        if (lane == 0) ssh[k] = acc;
    }
    __syncthreads();
    // block max
    float lmax = fmaxf(ssh[tid], ssh[tid + 256]);
    #pragma unroll
    for (int off = 16; off; off >>= 1) lmax = fmaxf(lmax, __shfl_xor(lmax, off, 32));
    if (lane == 0) red[wave] = lmax;
    __syncthreads();
    if (tid < 8) {
        float mm = red[tid];
        #pragma unroll
        for (int off = 4; off; off >>= 1) mm = fmaxf(mm, __shfl_xor(mm, off, 32));
        if (tid == 0) red[0] = mm;
    }
    __syncthreads();
    const float mx = red[0];
    const float e0 = fast_exp(ssh[tid] - mx);
    const float e1 = fast_exp(ssh[tid + 256] - mx);
    float lsum = e0 + e1;
    #pragma unroll
    for (int off = 16; off; off >>= 1) lsum += __shfl_xor(lsum, off, 32);
    __syncthreads();
    if (lane == 0) red[wave] = lsum;
    __syncthreads();
    if (tid < 8) {
        float s = red[tid];
        #pragma unroll
        for (int off = 4; off; off >>= 1) s += __shfl_xor(s, off, 32);
        if (tid == 0) red[0] = s;
    }
    __syncthreads();
    const float inv = 1.0f / red[0];
    rawS[(size_t)bq * 512 + tid]         = ssh[tid];
    rawS[(size_t)bq * 512 + tid + 256]   = ssh[tid + 256];
    outNorm[(size_t)bq * 512 + tid]       = e0 * inv;
    outNorm[(size_t)bq * 512 + tid + 256] = e1 * inv;
}

// ---------- kernel 3: context[b] = rawScores[b] (64x512) * keys[b] (512x512) ----------
// Block = 8 waves sharing one 16-row scores strip (staged once in LDS, 33 KB);
// B (keys) rows are contiguous in n -> coalesced global reads.
__global__ void context_wmma(const float* __restrict__ S,
                             const float* __restrict__ keys,
                             float* __restrict__ ctx) {
    __shared__ __align__(16) float As[16 * 516];   // 516 mod 64 = 4 -> conflict-free
    const int tid  = threadIdx.x;
    const int wave = tid >> 5;
    const int lane = tid & 31;
    const int b       = blockIdx.x >> 4;           // 8 batches x 4 rowTiles x 4 colGroups
    const int rowTile = (blockIdx.x >> 2) & 3;
    const int colTile = (blockIdx.x & 3) * 8 + wave;
    const float* A  = S    + (size_t)b * 64  * 512;
    const float* Bm = keys + (size_t)b * 512 * 512;
    const int m     = lane & 15;
    const int kbase = (lane >> 4) << 1;
    const int rowBase = rowTile * 16;
    const int colBase = colTile * 16;
    // stage scores strip: 16 rows x 512 floats = 2048 float4 slots
    for (int s = tid; s < 2048; s += 256) {
        const int r = s >> 7, c4 = s & 127;        // 128 float4 per row
        copy16_g2l(&As[r * 516 + c4 * 4],
                   A + (size_t)(rowBase + r) * 512 + c4 * 4);
    }
    async_join();
    __syncthreads();
    v8f acc = {};
    const float* arow = &As[m * 516];
    for (int k = 0; k < 512; k += 4) {
        v2f a = *(const v2f*)(arow + k + kbase);
        v2f bf;
        bf[0] = Bm[(size_t)(k + kbase)     * 512 + colBase + m];  // coalesced in n
        bf[1] = Bm[(size_t)(k + kbase + 1) * 512 + colBase + m];
        acc = __builtin_amdgcn_wmma_f32_16x16x4_f32(
            false, a, false, bf, (short)0, acc, false, false);
    }
    const int col  = lane & 15;
    const int rsel = (lane >> 4) * 8;
    #pragma unroll
    for (int j = 0; j < 8; ++j)
        ctx[(size_t)b * 64 * 512 + (size_t)(rowBase + rsel + j) * 512 + colBase + col] = acc[j];
}

extern "C" void kernel_launch(void* const* d_in, const int* in_sizes, int n_in,
                              void* d_out, int out_size, void* d_ws, size_t ws_size,
                              hipStream_t stream) {
    const float* query = (const float*)d_in[0];   // [8,64,512]
    const float* keys  = (const float*)d_in[1];   // [8,512,512]
    const float* Wq    = (const float*)d_in[2];   // [512,512]
    const float* Wk    = (const float*)d_in[3];   // [512,512]
    const float* la    = (const float*)d_in[4];   // [512]
    const float* nsc   = (const float*)d_in[5];   // [1]
    const float* nbias = (const float*)d_in[6];   // [512]

    float* ctx_out  = (float*)d_out;              // [8,64,512]
    float* norm_out = (float*)d_out + 8 * 64 * 512;

    float* ws   = (float*)d_ws;
    float* v    = ws;                      // 512
    float* pq   = v + 512;                 // 512*512   = 262144
    float* pk   = pq + 512 * 512;          // 4096*512  = 2097152
    float* rawS = pk + 4096 * 512;         // 512*512   = 262144

    compute_v_kernel<<<1, 512, 0, stream>>>(la, nsc, v);
    gemm_xWt_wmma<<<(512 / 16) * 4, 256, 0, stream>>>(query, Wq, pq);     // pq: 512 rows
    gemm_xWt_wmma<<<(4096 / 16) * 4, 256, 0, stream>>>(keys, Wk, pk);     // pk: 4096 rows
    scores_softmax_kernel<<<512, 256, 0, stream>>>(pq, pk, nbias, v, rawS, norm_out);
    context_wmma<<<128, 256, 0, stream>>>(rawS, keys, ctx_out);
}